// TreeNN_42477226557553
// MI455X (gfx1250) — compile-verified
//
#include <hip/hip_runtime.h>
#include <math.h>

typedef float v2f __attribute__((ext_vector_type(2)));
typedef float v8f __attribute__((ext_vector_type(8)));

#define DMODEL 256
#define KDIM 512              // 2*DMODEL
#define GOUT 1280             // 5*DMODEL
#define LDS_STRIDE 516        // 512 + 4 pad -> rows advance 4 banks, 16 rows cover all 64

__device__ __forceinline__ float sigf(float x) {
    return 1.0f / (1.0f + __expf(-x));
}

// ---------------------------------------------------------------------------
// Kernel A: W (512 x 1280, row-major) -> WT (1280 x 512) so that consecutive
// K for a fixed output column are contiguous (float2 B-fragment loads).
// ---------------------------------------------------------------------------
__global__ void __launch_bounds__(256) transpose_w_kernel(
        const float* __restrict__ W, float* __restrict__ WT) {
    int idx = blockIdx.x * blockDim.x + threadIdx.x;
    if (idx >= KDIM * GOUT) return;
    int k = idx / GOUT;
    int c = idx - k * GOUT;
    WT[(size_t)c * KDIM + k] = W[idx];   // coalesced read, strided write (L2 absorbs)
}

// ---------------------------------------------------------------------------
// Kernel B: leaf embedding with norm clipping. One wave32 per leaf row (256f).
// acts == d_out directly; mem rows of leaves are zeroed here.
// ---------------------------------------------------------------------------
__global__ void __launch_bounds__(256) leaf_embed_kernel(
        const int* __restrict__ tokens, const int* __restrict__ leaf_ids,
        const float* __restrict__ emb, float* __restrict__ acts,
        float* __restrict__ mem, int nleaf) {
    int wave = threadIdx.x >> 5;
    int lane = threadIdx.x & 31;
    int g = blockIdx.x * (blockDim.x >> 5) + wave;
    if (g >= nleaf) return;
    int lid = leaf_ids[g];
    int tok = tokens[lid];
    const float* e = emb + (size_t)tok * DMODEL + lane * 8;
    float4 v0 = *(const float4*)(e);
    float4 v1 = *(const float4*)(e + 4);
    float s = v0.x*v0.x + v0.y*v0.y + v0.z*v0.z + v0.w*v0.w
            + v1.x*v1.x + v1.y*v1.y + v1.z*v1.z + v1.w*v1.w;
    #pragma unroll
    for (int m = 16; m >= 1; m >>= 1) s += __shfl_xor(s, m, 32);
    float scale = fminf(1.0f, 1.0f / fmaxf(sqrtf(s), 1e-7f));
    float4 o0 = make_float4(v0.x*scale, v0.y*scale, v0.z*scale, v0.w*scale);
    float4 o1 = make_float4(v1.x*scale, v1.y*scale, v1.z*scale, v1.w*scale);
    float* arow = acts + (size_t)lid * DMODEL + lane * 8;
    *(float4*)(arow)     = o0;
    *(float4*)(arow + 4) = o1;
    float4 z = make_float4(0.f, 0.f, 0.f, 0.f);
    float* mrow = mem + (size_t)lid * DMODEL + lane * 8;
    *(float4*)(mrow)     = z;
    *(float4*)(mrow + 4) = z;
}

// ---------------------------------------------------------------------------
// Kernel C: one TreeLSTM level. Block = 128 threads (4 waves).
// Block handles 16 nodes; wave w handles output columns (blockIdx.y*4+w)*16.
// LDS A-panel: 16 rows x 512 K of concatenated [acts[left] | acts[right]].
// Each wave accumulates 5 gate tiles (i,fl,fr,o,u) of 16x16 with
// v_wmma_f32_16x16x4_f32 over K=512 (128 steps), then applies the cell.
// ---------------------------------------------------------------------------
__global__ void __launch_bounds__(128) tree_level_kernel(
        const int* __restrict__ internal_ids, const int* __restrict__ left_idx,
        const int* __restrict__ right_idx, const float* __restrict__ WT,
        const float* __restrict__ bias, float* __restrict__ acts,
        float* __restrict__ mem, int level_start) {
    __shared__ float sA[16 * LDS_STRIDE];
    __shared__ int sId[16], sLi[16], sRi[16];

    int tid = threadIdx.x;
    int base = blockIdx.x * 16;

    if (tid < 16) {
        int id = internal_ids[level_start + base + tid];
        sId[tid] = id;
        sLi[tid] = left_idx[id];
        sRi[tid] = right_idx[id];
    }
    __syncthreads();

    // Cooperative A-panel stage: 8 threads per node row, 16 float4 each.
    {
        int r = tid >> 3;
        const float* srcL = acts + (size_t)sLi[r] * DMODEL;
        const float* srcR = acts + (size_t)sRi[r] * DMODEL;
        float* dst = sA + r * LDS_STRIDE;
        #pragma unroll
        for (int j = 0; j < 16; ++j) {
            int q = (tid & 7) + j * 8;            // float4 index within 512-f row
            float4 v = (q < 64) ? *(const float4*)(srcL + q * 4)
                                : *(const float4*)(srcR + (q - 64) * 4);
            *(float4*)(dst + q * 4) = v;
        }
    }
    __syncthreads();

    int wave = tid >> 5;
    int lane = tid & 31;
    int colbase = (blockIdx.y * 4 + wave) * 16;   // 16 of 256 logical output dims
    int row  = lane & 15;                         // A-frag: M row
    int col  = lane & 15;                         // B/C-frag: N col
    int k0   = (lane >> 4) * 2;                   // K pair selector

    const float* ap = sA + row * LDS_STRIDE + k0;
    const float* w0 = WT + (size_t)(0 * DMODEL + colbase + col) * KDIM + k0;
    const float* w1 = WT + (size_t)(1 * DMODEL + colbase + col) * KDIM + k0;
    const float* w2 = WT + (size_t)(2 * DMODEL + colbase + col) * KDIM + k0;
    const float* w3 = WT + (size_t)(3 * DMODEL + colbase + col) * KDIM + k0;
    const float* w4 = WT + (size_t)(4 * DMODEL + colbase + col) * KDIM + k0;

    v8f acc0 = {}, acc1 = {}, acc2 = {}, acc3 = {}, acc4 = {};

    #pragma unroll 4
    for (int kk = 0; kk < KDIM; kk += 4) {
        v2f a  = *(const v2f*)(ap + kk);          // LDS, conflict-free via pad
        v2f b0 = *(const v2f*)(w0 + kk);
        v2f b1 = *(const v2f*)(w1 + kk);
        v2f b2 = *(const v2f*)(w2 + kk);
        v2f b3 = *(const v2f*)(w3 + kk);
        v2f b4 = *(const v2f*)(w4 + kk);
        acc0 = __builtin_amdgcn_wmma_f32_16x16x4_f32(false, a, false, b0, (short)0, acc0, false, false);
        acc1 = __builtin_amdgcn_wmma_f32_16x16x4_f32(false, a, false, b1, (short)0, acc1, false, false);
        acc2 = __builtin_amdgcn_wmma_f32_16x16x4_f32(false, a, false, b2, (short)0, acc2, false, false);
        acc3 = __builtin_amdgcn_wmma_f32_16x16x4_f32(false, a, false, b3, (short)0, acc3, false, false);
        acc4 = __builtin_amdgcn_wmma_f32_16x16x4_f32(false, a, false, b4, (short)0, acc4, false, false);
    }

    float bv0 = bias[0 * DMODEL + colbase + col];
    float bv1 = bias[1 * DMODEL + colbase + col];
    float bv2 = bias[2 * DMODEL + colbase + col];
    float bv3 = bias[3 * DMODEL + colbase + col];
    float bv4 = bias[4 * DMODEL + colbase + col];

    int rhalf = (lane >> 4) * 8;                  // C/D layout: VGPR p holds rows p, p+8
    #pragma unroll
    for (int p = 0; p < 8; ++p) {
        int r = rhalf + p;
        float iv = acc0[p] + bv0;
        float fl = acc1[p] + bv1;
        float fr = acc2[p] + bv2;
        float ov = acc3[p] + bv3;
        float uv = acc4[p] + bv4;
        float cl = mem[(size_t)sLi[r] * DMODEL + colbase + col];
        float cr = mem[(size_t)sRi[r] * DMODEL + colbase + col];
        float c  = sigf(iv) * tanhf(uv) + sigf(fl) * cl + sigf(fr) * cr;
        float h  = sigf(ov) * tanhf(c);
        size_t o = (size_t)sId[r] * DMODEL + colbase + col;
        acts[o] = h;
        mem[o]  = c;
    }
}

// ---------------------------------------------------------------------------
// Launch: transpose W, embed leaves, then 11 level kernels (launch order gives
// the inter-level barrier). mem + WT live in d_ws; acts is d_out directly.
// ---------------------------------------------------------------------------
extern "C" void kernel_launch(void* const* d_in, const int* in_sizes, int n_in,
                              void* d_out, int out_size, void* d_ws, size_t ws_size,
                              hipStream_t stream) {
    const int*   tokens       = (const int*)d_in[0];
    const int*   left_idx     = (const int*)d_in[1];
    const int*   right_idx    = (const int*)d_in[2];
    // d_in[3]=depths, d_in[4]=operation_order, d_in[5]=lengths: structural, unused
    const int*   leaf_ids     = (const int*)d_in[6];
    const int*   internal_ids = (const int*)d_in[7];
    const float* emb          = (const float*)d_in[8];
    const float* W            = (const float*)d_in[9];
    const float* bias         = (const float*)d_in[10];

    float* acts = (float*)d_out;
    const int N = in_sizes[0];                     // 65520 nodes
    float* mem = (float*)d_ws;                     // N * 256 floats
    float* WT  = mem + (size_t)N * DMODEL;         // 1280 * 512 floats

    // A: W transpose (fully L2-resident afterwards)
    int nW = KDIM * GOUT;
    transpose_w_kernel<<<(nW + 255) / 256, 256, 0, stream>>>(W, WT);

    // B: leaves (one wave per leaf)
    int nleaf = in_sizes[6];                       // 32768
    leaf_embed_kernel<<<(nleaf + 7) / 8, 256, 0, stream>>>(
        tokens, leaf_ids, emb, acts, mem, nleaf);

    // C: levels 1..11, bottom-up
    const int DEPTH = 11, BTREES = 16;
    int start = 0;
    for (int k = 1; k <= DEPTH; ++k) {
        int width  = BTREES << (DEPTH - k);        // nodes at this level
        int ntiles = width / 16;                   // >= 1 (level 11: exactly 1)
        dim3 grid(ntiles, 4);
        tree_level_kernel<<<grid, 128, 0, stream>>>(
            internal_ids, left_idx, right_idx, WT, bias, acts, mem, start);
        start += width;
    }
}